// RBFKernelLayer_80719615361567
// MI455X (gfx1250) — compile-verified
//
#include <hip/hip_runtime.h>

// Problem constants (from reference): B=4096, IN=512, OUT=512, NC=64, S=1
#define B_ROWS 4096
#define IN_F   512
#define OUT_F  512
#define NC_    64

// IN-reduction split for the HBM-bound streaming phase
#define NCHUNK 16
#define I_PER_CHUNK (IN_F / NCHUNK)      // 32
#define ELEMS (OUT_F * NC_)              // 32768 (o,n) elements

typedef float v2f __attribute__((ext_vector_type(2)));
typedef float v4f __attribute__((ext_vector_type(4)));
typedef float v8f __attribute__((ext_vector_type(8)));

// ---------------------------------------------------------------------------
// Kernel 1: stream the three (IN, OUT, NC) tensors once (192 MB -> HBM bound).
// 2D grid: x = (o,n) slabs (each thread owns 4 contiguous elements, b128
// loads), y = IN chunk (16 chunks x 32 i-steps). 512 blocks / 4096 waves of
// in-flight b128 traffic to saturate 23.3 TB/s. Partial sums go to workspace;
// a finalize pass folds the 16 chunks deterministically.
// ---------------------------------------------------------------------------
__global__ void reduce_coeffs_kernel(const float* __restrict__ cm,
                                     const float* __restrict__ clv,
                                     const float* __restrict__ eps,
                                     float* __restrict__ c_partial,  // [NCHUNK][ELEMS]
                                     float* __restrict__ klpart) {   // [32*NCHUNK]
  const int e4    = blockIdx.x * 256 + threadIdx.x;   // which float4 (0..8191)
  const int chunk = blockIdx.y;
  const size_t base = (size_t)chunk * I_PER_CHUNK * ELEMS + (size_t)e4 * 4;

  v4f s  = {};
  float kl = 0.f;
  size_t p = base;
  for (int i = 0; i < I_PER_CHUNK; ++i, p += ELEMS) {
    v4f m  = *reinterpret_cast<const v4f*>(cm  + p);
    v4f lv = *reinterpret_cast<const v4f*>(clv + p);
    v4f e  = *reinterpret_cast<const v4f*>(eps + p);
#pragma unroll
    for (int j = 0; j < 4; ++j) {
      s[j] += m[j] + e[j] * __expf(0.5f * lv[j]);        // reparameterized coeff
      kl   += __expf(lv[j]) + m[j] * m[j] - 1.0f - lv[j]; // KL integrand (x2)
    }
  }
  *reinterpret_cast<v4f*>(c_partial + (size_t)chunk * ELEMS + (size_t)e4 * 4) = s;

  __shared__ float red[256];
  red[threadIdx.x] = kl;
  __syncthreads();
  for (int o = 128; o > 0; o >>= 1) {
    if (threadIdx.x < o) red[threadIdx.x] += red[threadIdx.x + o];
    __syncthreads();
  }
  if (threadIdx.x == 0) klpart[blockIdx.y * gridDim.x + blockIdx.x] = red[0];
}

// Fold the NCHUNK partial c_sum arrays into the final c_sum (S==1 -> no /S).
__global__ void csum_final_kernel(const float* __restrict__ c_partial,
                                  float* __restrict__ c_sum) {
  const int idx = blockIdx.x * 256 + threadIdx.x;   // 0..ELEMS-1
  float s = 0.f;
#pragma unroll
  for (int c = 0; c < NCHUNK; ++c) s += c_partial[(size_t)c * ELEMS + idx];
  c_sum[idx] = s;
}

__global__ void kl_final_kernel(const float* __restrict__ klpart,
                                float* __restrict__ out_kl) {
  __shared__ float red[256];
  float s = klpart[threadIdx.x] + klpart[threadIdx.x + 256];  // 512 partials
  red[threadIdx.x] = s;
  __syncthreads();
  for (int o = 128; o > 0; o >>= 1) {
    if (threadIdx.x < o) red[threadIdx.x] += red[threadIdx.x + o];
    __syncthreads();
  }
  if (threadIdx.x == 0) *out_kl = 0.5f * red[0];
}

// ---------------------------------------------------------------------------
// Kernel 2: per-center ||c||^2 and 1/(2*sigma^2 + 1e-8)
// ---------------------------------------------------------------------------
__global__ void center_norm_kernel(const float* __restrict__ centers,
                                   const float* __restrict__ log_sigma,
                                   float* __restrict__ cnorm,
                                   float* __restrict__ invden) {
  const int n = blockIdx.x;
  __shared__ float red[128];
  float s = 0.f;
  for (int i = threadIdx.x; i < IN_F; i += 128) {
    float v = centers[(size_t)n * IN_F + i];
    s += v * v;
  }
  red[threadIdx.x] = s;
  __syncthreads();
  for (int o = 64; o > 0; o >>= 1) {
    if (threadIdx.x < o) red[threadIdx.x] += red[threadIdx.x + o];
    __syncthreads();
  }
  if (threadIdx.x == 0) {
    cnorm[n] = red[0];
    float sg = __expf(log_sigma[n]);
    invden[n] = 1.0f / (2.0f * sg * sg + 1e-8f);
  }
}

// ---------------------------------------------------------------------------
// Kernel 3: per-row ||x||^2
// ---------------------------------------------------------------------------
__global__ void x_norm_kernel(const float* __restrict__ x,
                              float* __restrict__ xnorm) {
  const int b = blockIdx.x;
  __shared__ float red[128];
  float s = 0.f;
  for (int i = threadIdx.x; i < IN_F; i += 128) {
    float v = x[(size_t)b * IN_F + i];
    s += v * v;
  }
  red[threadIdx.x] = s;
  __syncthreads();
  for (int o = 64; o > 0; o >>= 1) {
    if (threadIdx.x < o) red[threadIdx.x] += red[threadIdx.x + o];
    __syncthreads();
  }
  if (threadIdx.x == 0) xnorm[b] = red[0];
}

// ---------------------------------------------------------------------------
// Kernel 4: RBF via WMMA f32 16x16x4.
// G = x @ centers^T (M=4096, N=64, K=512), then
// rbf[b,n] = exp(-(||x_b||^2 - 2G + ||c_n||^2) * invden[n]).
// A fragment (16x4 f32): lane L holds row M=L&15, float2 = K = 2*(L>>4)+{0,1}.
// B fragment (4x16 f32): lane L holds col N=L&15, float2 = K = 2*(L>>4)+{0,1}.
// C/D (16x16 f32): VGPR r -> M = r + 8*(L>>4), N = L&15.
// ---------------------------------------------------------------------------
__global__ void rbf_kernel(const float* __restrict__ x,
                           const float* __restrict__ centers,
                           const float* __restrict__ xnorm,
                           const float* __restrict__ cnorm,
                           const float* __restrict__ invden,
                           float* __restrict__ rbf) {
  const int lane = threadIdx.x & 31;
  const int wave = threadIdx.x >> 5;
  const int rowBase = blockIdx.y * 128 + wave * 16;
  const int colBase = blockIdx.x * 16;
  const int mrow = lane & 15;
  const int koff = (lane >> 4) * 2;

  const float* arow = x       + (size_t)(rowBase + mrow) * IN_F + koff;
  const float* brow = centers + (size_t)(colBase + mrow) * IN_F + koff;

  v8f c = {};
#pragma unroll 4
  for (int k = 0; k < IN_F; k += 4) {
    v2f a = *reinterpret_cast<const v2f*>(arow + k);
    v2f b = *reinterpret_cast<const v2f*>(brow + k);
    c = __builtin_amdgcn_wmma_f32_16x16x4_f32(false, a, false, b,
                                              (short)0, c, false, false);
  }

  const int half = lane >> 4;
  const int col = colBase + mrow;
  const float inv = invden[col];
  const float cn  = cnorm[col];
#pragma unroll
  for (int r = 0; r < 8; ++r) {
    int row = rowBase + r + 8 * half;
    float d = xnorm[row] - 2.0f * c[r] + cn;
    rbf[(size_t)row * NC_ + col] = __expf(-d * inv);
  }
}

// ---------------------------------------------------------------------------
// Kernel 5: fused output GEMM via WMMA f32 16x16x4.
// out = rbf @ c_sum^T  (K=NC=64)  +  x @ base_weight  (K=IN=512)
// One 16x16 output tile per wave; 8 waves/block -> 128x16 macro-tile.
// All operands are L2-resident (x 8MB, W 1MB, rbf 1MB, c_sum 128KB).
// ---------------------------------------------------------------------------
__global__ void out_gemm_kernel(const float* __restrict__ x,
                                const float* __restrict__ w,
                                const float* __restrict__ rbf,
                                const float* __restrict__ c_sum,
                                float* __restrict__ out) {
  const int lane = threadIdx.x & 31;
  const int wave = threadIdx.x >> 5;
  const int rowBase = blockIdx.y * 128 + wave * 16;
  const int colBase = blockIdx.x * 16;
  const int mrow = lane & 15;
  const int koff = (lane >> 4) * 2;
  const int col = colBase + mrow;

  v8f c = {};

  // Phase 1: rbf @ c_sum^T, B[k][n] = c_sum[n*NC + k] (contiguous float2)
  {
    const float* arow = rbf   + (size_t)(rowBase + mrow) * NC_ + koff;
    const float* brow = c_sum + (size_t)col * NC_ + koff;
#pragma unroll
    for (int k = 0; k < NC_; k += 4) {
      v2f a = *reinterpret_cast<const v2f*>(arow + k);
      v2f b = *reinterpret_cast<const v2f*>(brow + k);
      c = __builtin_amdgcn_wmma_f32_16x16x4_f32(false, a, false, b,
                                                (short)0, c, false, false);
    }
  }

  // Phase 2: x @ W, W row-major (IN, OUT): B[k][n] = w[k*OUT + n]
  {
    const float* arow = x + (size_t)(rowBase + mrow) * IN_F + koff;
#pragma unroll 4
    for (int k = 0; k < IN_F; k += 4) {
      v2f a = *reinterpret_cast<const v2f*>(arow + k);
      v2f b;
      b.x = w[(size_t)(k + koff)     * OUT_F + col];
      b.y = w[(size_t)(k + koff + 1) * OUT_F + col];
      c = __builtin_amdgcn_wmma_f32_16x16x4_f32(false, a, false, b,
                                                (short)0, c, false, false);
    }
  }

  const int half = lane >> 4;
#pragma unroll
  for (int r = 0; r < 8; ++r) {
    int row = rowBase + r + 8 * half;
    out[(size_t)row * OUT_F + col] = c[r];
  }
}

// ---------------------------------------------------------------------------
extern "C" void kernel_launch(void* const* d_in, const int* in_sizes, int n_in,
                              void* d_out, int out_size, void* d_ws, size_t ws_size,
                              hipStream_t stream) {
  (void)in_sizes; (void)n_in; (void)out_size; (void)ws_size;

  const float* x             = (const float*)d_in[0];
  const float* centers       = (const float*)d_in[1];
  const float* log_sigma     = (const float*)d_in[2];
  const float* coeff_mean    = (const float*)d_in[3];
  const float* coeff_log_var = (const float*)d_in[4];
  const float* base_weight   = (const float*)d_in[5];
  const float* eps           = (const float*)d_in[6];
  float* out = (float*)d_out;                    // [B*OUT floats] ++ [kl]

  // Workspace layout (floats): ~3.4 MB total
  float* ws        = (float*)d_ws;
  float* c_partial = ws;                                   // NCHUNK*ELEMS = 524288
  float* c_sum     = c_partial + (size_t)NCHUNK * ELEMS;   // ELEMS        = 32768
  float* rbf       = c_sum  + ELEMS;                       // B*NC         = 262144
  float* xnorm     = rbf    + (size_t)B_ROWS * NC_;        // B            = 4096
  float* cnorm     = xnorm  + B_ROWS;                      // NC           = 64
  float* invden    = cnorm  + NC_;                         // NC           = 64
  float* klp       = invden + NC_;                         // 512 partials

  // 1) Stream 192 MB once with b128 loads, 512 blocks (HBM-bound phase)
  reduce_coeffs_kernel<<<dim3(ELEMS / (256 * 4), NCHUNK), dim3(256), 0, stream>>>(
      coeff_mean, coeff_log_var, eps, c_partial, klp);

  // 2) Fold chunk partials into final c_sum (deterministic order)
  csum_final_kernel<<<dim3(ELEMS / 256), dim3(256), 0, stream>>>(c_partial, c_sum);

  // 3) KL scalar (deterministic tree reduce, no float atomics)
  kl_final_kernel<<<dim3(1), dim3(256), 0, stream>>>(
      klp, out + (size_t)B_ROWS * OUT_F);

  // 4) Center norms + 1/(2 sigma^2 + 1e-8)
  center_norm_kernel<<<dim3(NC_), dim3(128), 0, stream>>>(
      centers, log_sigma, cnorm, invden);

  // 5) Row norms of x
  x_norm_kernel<<<dim3(B_ROWS), dim3(128), 0, stream>>>(x, xnorm);

  // 6) RBF basis via f32 WMMA (x @ centers^T, then exp)
  rbf_kernel<<<dim3(NC_ / 16, B_ROWS / 128), dim3(256), 0, stream>>>(
      x, centers, xnorm, cnorm, invden, rbf);

  // 7) Fused output GEMM via f32 WMMA: rbf @ c_sum^T + x @ base_weight
  out_gemm_kernel<<<dim3(OUT_F / 16, B_ROWS / 128), dim3(256), 0, stream>>>(
      x, base_weight, rbf, c_sum, out);
}